// CosFace_norm_26336739459514
// MI455X (gfx1250) — compile-verified
//
#include <hip/hip_runtime.h>
#include <hip/hip_bf16.h>
#include <math.h>

#define SCALE_S 64.0f
#define MARGIN_M 0.35f

#define TILE_OUT 2040   // outputs per block
#define TILE_IN  2048   // LDS staging floats (covers TILE_OUT+1 shifted cols + align slop)
#define BLOCK    256

typedef int v4i __attribute__((ext_vector_type(4)));
#define AS1 __attribute__((address_space(1)))
#define AS3 __attribute__((address_space(3)))

#if defined(__gfx1250__) && __has_builtin(__builtin_amdgcn_global_load_async_to_lds_b128)
#define HAVE_ASYNC_LDS 1
#else
#define HAVE_ASYNC_LDS 0
#endif

#if defined(__gfx1250__) && __has_builtin(__builtin_amdgcn_s_wait_asynccnt)
#define WAIT_ASYNC() __builtin_amdgcn_s_wait_asynccnt(0)
#else
#define WAIT_ASYNC() asm volatile("s_wait_asynccnt 0" ::: "memory")
#endif

// ---------------------------------------------------------------------------
// Kernel: diff[r, j] = S * (logits[r, j + (j>=L)] - (target - M))
// Streams one 2040-wide output tile per block through an 8KB LDS window
// filled with GLOBAL_LOAD_ASYNC_TO_LDS_B128 (CDNA5 async data path).
// Output stores are non-temporal (write-once stream, never re-read).
// ---------------------------------------------------------------------------
__global__ __launch_bounds__(BLOCK) void cosface_diff_kernel(
    const float* __restrict__ logits, const int* __restrict__ labels,
    float* __restrict__ out, int B, int C) {
  __shared__ float tile[TILE_IN];

  const int r   = blockIdx.y;
  const int t0  = blockIdx.x * TILE_OUT;          // first output col of tile
  const int tid = threadIdx.x;
  const int jend = C - 1;                         // outputs per row

  const int L = labels[r];                        // block-uniform -> s_load
  const size_t rowg  = (size_t)r * (size_t)C;
  const size_t total = (size_t)B * (size_t)C;

  // Input window: cols [base, base + TILE_IN). base = align4(t0).
  // Needed cols are [t0, t0 + TILE_OUT] minus {L}; span <= TILE_OUT + 4 <= TILE_IN.
  const int base = t0 & ~3;

  // Stage input window into LDS: 2 x b128 per thread = 2048 floats.
#pragma unroll
  for (int k = 0; k < TILE_IN / (BLOCK * 4); ++k) {
    const int li = (tid + k * BLOCK) * 4;         // float index in LDS
    size_t g = rowg + (size_t)(base + li);
    if (g > total - 4) g = total - 4;             // clamp to buffer end (garbage unused)
#if HAVE_ASYNC_LDS
    __builtin_amdgcn_global_load_async_to_lds_b128(
        (AS1 v4i*)(logits + g), (AS3 v4i*)(&tile[li]), 0, 0);
#else
    float4 v = *(const float4*)(logits + g);
    *(float4*)(&tile[li]) = v;
#endif
  }

  // Block-uniform target bias: S * (target_logit - M)
  const float sf = SCALE_S * (logits[rowg + (size_t)L] - MARGIN_M);

#if HAVE_ASYNC_LDS
  WAIT_ASYNC();
#endif
  __syncthreads();

  float* __restrict__ orow = out + (size_t)r * (size_t)jend;
#pragma unroll
  for (int k = 0; k < (TILE_OUT + BLOCK - 1) / BLOCK; ++k) {
    const int jl = tid + k * BLOCK;               // monotonic in k
    if (jl >= TILE_OUT) break;
    const int j = t0 + jl;
    if (j >= jend) break;
    const int col = j + (j >= L);                 // branchless skip of label column
    const float x = tile[col - base];
    __builtin_nontemporal_store(fmaf(SCALE_S, x, -sf), &orow[j]);
  }
}

// ---------------------------------------------------------------------------
// Kernel: per-row scalars. sin_theta, sin_theta_plus_m, sin_m  (B each)
// ---------------------------------------------------------------------------
__global__ __launch_bounds__(256) void cosface_row_kernel(
    const float* __restrict__ logits, const int* __restrict__ labels,
    float* __restrict__ sins, int B, int C) {
  int r = blockIdx.x * blockDim.x + threadIdx.x;
  if (r >= B) return;
  int L = labels[r];
  float x = logits[(size_t)r * (size_t)C + (size_t)L];
  float theta = acosf(x);
  float theta_m = acosf(x - MARGIN_M);
  sins[r]         = sinf(theta);            // sin_theta
  sins[B + r]     = sinf(theta_m);          // sin_theta_plus_m
  sins[2 * B + r] = sinf(theta_m - theta);  // sin_m
}

// ---------------------------------------------------------------------------
extern "C" void kernel_launch(void* const* d_in, const int* in_sizes, int n_in,
                              void* d_out, int out_size, void* d_ws, size_t ws_size,
                              hipStream_t stream) {
  const float* logits = (const float*)d_in[0];
  const int*   labels = (const int*)d_in[1];
  float* out = (float*)d_out;

  const int B = in_sizes[1];
  const int C = in_sizes[0] / B;

  float* diff = out;                                  // [B, C-1]
  float* sins = out + (size_t)B * (size_t)(C - 1);    // 3 x [B]

  const int tiles = (C - 1 + TILE_OUT - 1) / TILE_OUT;
  cosface_diff_kernel<<<dim3(tiles, B), BLOCK, 0, stream>>>(logits, labels, diff, B, C);

  cosface_row_kernel<<<(B + 255) / 256, 256, 0, stream>>>(logits, labels, sins, B, C);
}